// Projection_25099788877928
// MI455X (gfx1250) — compile-verified
//
#include <hip/hip_runtime.h>

#define NUM_REL 500
#define BATCH   4096
#define D       256
#define APAD    4
#define ASTRIDE (D + APAD)

typedef __attribute__((ext_vector_type(2))) float        v2f;
typedef __attribute__((ext_vector_type(8))) float        v8f;
typedef __attribute__((ext_vector_type(4))) float        v4f;
typedef __attribute__((ext_vector_type(4))) unsigned int v4u;
typedef __attribute__((ext_vector_type(8))) int          v8i;
typedef __attribute__((ext_vector_type(4))) int          v4i;

// ---------------- binning kernels ----------------

__global__ void zero_ws_kernel(int* counts, int* cursors) {
    int t = blockIdx.x * blockDim.x + threadIdx.x;
    if (t < NUM_REL) { counts[t] = 0; cursors[t] = 0; }
}

__global__ void count_kernel(const int* __restrict__ rel, int* __restrict__ counts) {
    int b = blockIdx.x * blockDim.x + threadIdx.x;
    if (b < BATCH) atomicAdd(&counts[rel[b]], 1);
}

__global__ void scan_kernel(const int* __restrict__ counts, int* __restrict__ offsets) {
    if (blockIdx.x == 0 && threadIdx.x == 0) {
        int acc = 0;
        for (int i = 0; i < NUM_REL; ++i) { offsets[i] = acc; acc += counts[i]; }
        offsets[NUM_REL] = acc;
    }
}

__global__ void scatter_kernel(const int* __restrict__ rel,
                               const int* __restrict__ offsets,
                               int* __restrict__ cursors,
                               int* __restrict__ bucket) {
    int b = blockIdx.x * blockDim.x + threadIdx.x;
    if (b < BATCH) {
        int r = rel[b];
        int pos = atomicAdd(&cursors[r], 1);
        bucket[offsets[r] + pos] = b;
    }
}

// ---------------- grouped GEMM: TDM gather -> LDS, V_WMMA_F32_16X16X4_F32 ----------------

__global__ __launch_bounds__(256)
void rel_gemm_kernel(const float* __restrict__ emb,
                     const float* __restrict__ mats,
                     const int*   __restrict__ counts,
                     const int*   __restrict__ offsets,
                     const int*   __restrict__ bucket,
                     float*       __restrict__ out) {
    __shared__ float Alds[16 * ASTRIDE];
    __shared__ int   rows[16];

    const int r   = blockIdx.x;
    const int n_r = counts[r];
    if (n_r == 0) return;
    const int off = offsets[r];
    const float* __restrict__ R = mats + (size_t)r * D * D;

    const int tid   = threadIdx.x;
    const int lane  = tid & 31;
    const int wave  = tid >> 5;
    const int m     = lane & 15;                 // A row within tile
    const int khalf = (lane >> 4) << 1;          // 0 or 2: K-pair base
    const int n0    = (wave * 2) * 16 + (lane & 15);
    const int n1    = n0 + 16;

    for (int m0 = 0; m0 < n_r; m0 += 16) {
        __syncthreads();                         // protect prev iteration's LDS reads
        if (tid < 16) rows[tid] = (m0 + tid < n_r) ? bucket[off + m0 + tid] : -1;

        const int nrows = (n_r - m0 < 16) ? (n_r - m0) : 16;

#if __has_builtin(__builtin_amdgcn_tensor_load_to_lds)
        // TDM gather mode: DMA `nrows` embedding rows (16-bit row indices) into
        // LDS, hardware-padding +4 DWORDs after every 256 DWORDs so the LDS
        // tile lands with a bank-conflict-free 260-float row stride.
        if (wave == 0) {
            unsigned int idx[16];
            #pragma unroll
            for (int i = 0; i < 16; ++i)
                idx[i] = (m0 + i < n_r) ? (unsigned int)bucket[off + m0 + i] : 0u;

            const unsigned long long gaddr = (unsigned long long)(uintptr_t)emb;
            const unsigned int lds_addr = (unsigned int)(uintptr_t)(&Alds[0]);

            v4u g0;
            g0.x = 1u | (1u << 31);                      // count=1, gather_mode=1, 16-bit idx
            g0.y = lds_addr;                             // lds_addr[31:0]
            g0.z = (unsigned int)gaddr;                  // global_addr[31:0]
            g0.w = (unsigned int)((gaddr >> 32) & 0x1FFFFFFu) | (2u << 30); // addr[56:32], type=2

            v8i g1;
            g1[0] = (2 << 16)        // data_size = 4B
                  | (1 << 20)        // pad_enable
                  | (7 << 22)        // pad_interval: 256 DWORDs (one 1024B row)
                  | (3 << 25);       // pad_amount: 4 DWORDs (16B)
            g1[1] = (D & 0xFFFF) << 16;                  // tensor_dim0[15:0]=256
            g1[2] = (BATCH & 0xFFFF) << 16;              // dim0[31:16]=0 | tensor_dim1[15:0]
            g1[3] = (D & 0xFFFF) << 16;                  // dim1[31:16]=0 | tile_dim0=256
            g1[4] = nrows;                               // tile_dim1 = #valid indices
            g1[5] = D;                                   // tensor_dim0_stride[31:0] = 256
            g1[6] = 0;
            g1[7] = 0;

            v4i g2, g3;
            #pragma unroll
            for (int j = 0; j < 4; ++j) {
                g2[j] = (int)(idx[2 * j]     | (idx[2 * j + 1] << 16));
                g3[j] = (int)(idx[8 + 2 * j] | (idx[9 + 2 * j] << 16));
            }

            v8i g4 = {};                                 // unused second-descriptor block

            __builtin_amdgcn_tensor_load_to_lds(g0, g1, g2, g3, g4, 0);
            __builtin_amdgcn_s_wait_tensorcnt(0);
        }
        __syncthreads();
#else
        __syncthreads();
        {   // manual staging fallback
            const int row  = tid >> 4;
            const int colb = (tid & 15) * 16;
            const int bidx = rows[row];
            float* dst = &Alds[row * ASTRIDE + colb];
            if (bidx >= 0) {
                const float* src = emb + (size_t)bidx * D + colb;
                #pragma unroll
                for (int j = 0; j < 4; ++j)
                    *(v4f*)(dst + 4 * j) = *(const v4f*)(src + 4 * j);
            } else {
                #pragma unroll
                for (int j = 0; j < 16; ++j) dst[j] = 0.0f;
            }
        }
        __syncthreads();
#endif

        v8f c0 = {};
        v8f c1 = {};
        const float* __restrict__ arow = &Alds[m * ASTRIDE];

        #pragma unroll 4
        for (int k = 0; k < D; k += 4) {
            const int kb = k + khalf;
            v2f a;
            a.x = arow[kb];
            a.y = arow[kb + 1];
            v2f b0, b1;
            b0.x = R[(size_t)kb * D + n0];
            b0.y = R[(size_t)(kb + 1) * D + n0];
            b1.x = R[(size_t)kb * D + n1];
            b1.y = R[(size_t)(kb + 1) * D + n1];
            c0 = __builtin_amdgcn_wmma_f32_16x16x4_f32(
                     false, a, false, b0, (short)0, c0, false, false);
            c1 = __builtin_amdgcn_wmma_f32_16x16x4_f32(
                     false, a, false, b1, (short)0, c1, false, false);
        }

        // Scatter C: vgpr i holds rows i (lanes 0-15) and i+8 (lanes 16-31).
        #pragma unroll
        for (int i = 0; i < 8; ++i) {
            const int row  = i + ((lane >> 4) << 3);
            const int bidx = rows[row];
            if (bidx >= 0) {
                out[(size_t)bidx * D + n0] = c0[i];
                out[(size_t)bidx * D + n1] = c1[i];
            }
        }
    }
}

// ---------------- launcher ----------------

extern "C" void kernel_launch(void* const* d_in, const int* in_sizes, int n_in,
                              void* d_out, int out_size, void* d_ws, size_t ws_size,
                              hipStream_t stream) {
    const float* emb  = (const float*)d_in[0];   // [4096, 1, 256] f32
    const int*   rel  = (const int*)d_in[1];     // [4096] int
    const float* mats = (const float*)d_in[2];   // [500, 256, 256] f32
    float*       out  = (float*)d_out;           // [4096, 1, 256] f32

    int* counts  = (int*)d_ws;          // 512 ints
    int* offsets = counts + 512;        // 512 ints (uses 501)
    int* cursors = offsets + 512;       // 512 ints
    int* bucket  = cursors + 512;       // 4096 ints

    zero_ws_kernel<<<(NUM_REL + 255) / 256, 256, 0, stream>>>(counts, cursors);
    count_kernel<<<(BATCH + 255) / 256, 256, 0, stream>>>(rel, counts);
    scan_kernel<<<1, 32, 0, stream>>>(counts, offsets);
    scatter_kernel<<<(BATCH + 255) / 256, 256, 0, stream>>>(rel, offsets, cursors, bucket);
    rel_gemm_kernel<<<NUM_REL, 256, 0, stream>>>(emb, mats, counts, offsets, bucket, out);
}